// DiffusionODEFunc_4320737100359
// MI455X (gfx1250) — compile-verified
//
#include <hip/hip_runtime.h>
#include <math.h>

// Problem constants (from reference): B=4, C=1, H=W=2048, cv/ch 1024^2,
// pooled 256^2, SCALE=8, L=0.24, EPS=1e-8.
#define HH      2048
#define WW      2048
#define CVN     1024
#define PS      256
#define TROWS   16         // tile rows (2 rows per thread, 2 pooled rows)
#define COLSZ   256        // tile cols (8 cols per thread)
#define PITCH   264        // LDS row pitch in floats (66 float4 chunks)
#define NCHK    66         // float4 chunks staged per row
#define NROWST  18         // staged y rows (tileR-1 .. tileR+16)
#define NSEG    132        // ch segment taps per row
#define NTHR    256
#define LCONST  0.24f
#define EPSV    1e-8f

__global__ __launch_bounds__(NTHR)
void diffusion_fused_kernel(const float* __restrict__ tptr,
                            const float* __restrict__ y,
                            const float* __restrict__ cv,
                            const float* __restrict__ ch,
                            const float* __restrict__ source,
                            const int*   __restrict__ mask_inv,
                            float*       __restrict__ out)
{
    // y tile: 18 rows x 264 staged cols (global cols tileC-4 .. tileC+259)
    __shared__ float ytile[NROWST * PITCH];
    // L-prescaled, vertically-lerped ch taps per output row of the tile
    __shared__ float chseg[TROWS * NSEG];
    __shared__ float psum[NTHR];
    __shared__ float rm1w[64];     // weight*(ratio-1) per 8x8 pooled block

    const int tid   = threadIdx.x;
    const int tileC = blockIdx.x * COLSZ;
    const int tileR = blockIdx.y * TROWS;
    const int bb    = blockIdx.z;

    const float* yb  = y  + (size_t)bb * HH * WW;
    const float* cvb = cv + (size_t)bb * CVN * CVN;
    const float* chb = ch + (size_t)bb * CVN * CVN;

    const float SV = 1024.0f / 2047.0f;

    // ---------------- async stage of y tile into LDS (CDNA5 async path) ----
    // 18 rows * 66 float4 chunks = 1188 b128 transfers; addresses clamped.
    // Clamped halo values only feed flux terms that are zeroed by guards.
    for (int idx = tid; idx < NROWST * NCHK; idx += NTHR) {
        int lrow = idx / NCHK;
        int q    = idx - lrow * NCHK;
        int gr   = tileR - 1 + lrow;
        gr = min(max(gr, 0), HH - 1);
        int gc   = tileC - 4 + 4 * q;
        gc = min(max(gc, 0), WW - 4);
        unsigned long long gaddr =
            (unsigned long long)(const void*)(yb + (size_t)gr * WW + gc);
        unsigned lds_off =
            (unsigned)(unsigned long long)(const void*)&ytile[lrow * PITCH + 4 * q];
        asm volatile("global_load_async_to_lds_b128 %0, %1, off"
                     :: "v"(lds_off), "v"(gaddr) : "memory");
    }

    // ---------------- cooperative ch_r segment (overlaps async latency) ----
    // ch_r = resize(ch,(2048,2047)): rows exact x2 (0.5*i-0.25), cols
    // u=(c+0.5)*1024/2047-0.5. Stage L * vertical-lerp(ch) at source cols
    // kLo..kLo+131 for each of the 16 tile rows.
    float uLo = ((float)(tileC - 1) + 0.5f) * SV - 0.5f;
    const int kLo = (int)floorf(uLo);
    for (int idx = tid; idx < TROWS * NSEG; idx += NTHR) {
        int ty2 = idx & 15;            // row in tile
        int kk  = idx >> 4;            // tap index 0..131
        int i2  = tileR + ty2;
        float xh = 0.5f * (float)i2 - 0.25f;
        float xf = floorf(xh);
        float wh = xh - xf;
        int h0  = (int)xf;
        int h0c = min(max(h0, 0), CVN - 1);
        int h1c = min(max(h0 + 1, 0), CVN - 1);
        int kcl = min(max(kLo + kk, 0), CVN - 1);
        float c0v = chb[(size_t)h0c * CVN + kcl];
        float c1v = chb[(size_t)h1c * CVN + kcl];
        chseg[ty2 * NSEG + kk] = LCONST * (c0v + wh * (c1v - c0v));
    }

    asm volatile("s_wait_asynccnt 0" ::: "memory");
    __syncthreads();

    // ---------------- per-thread geometry --------------------------------
    const int tx  = tid & 31;          // output cols j0..j0+7
    const int ty  = tid >> 5;          // row pair index (rows 2ty, 2ty+1)
    const int iA  = tileR + 2 * ty;    // first output row
    const int iB  = iA + 1;            // second output row
    const int j0  = tileC + 8 * tx;    // multiple of 8
    const int lrA = 2 * ty + 1;        // LDS row of iA
    const int lb  = 8 * tx;            // LDS col of (j0 - 4)

    // Four stencil rows iA-1..iA+2, cols j0-4 .. j0+11 (we use idx 3..12)
    float yA[16], yB[16], yC[16], yD[16];
    #pragma unroll
    for (int w = 0; w < 4; ++w) {
        float4 va = *(const float4*)&ytile[(lrA - 1) * PITCH + lb + 4 * w];
        float4 vb = *(const float4*)&ytile[(lrA    ) * PITCH + lb + 4 * w];
        float4 vc = *(const float4*)&ytile[(lrA + 1) * PITCH + lb + 4 * w];
        float4 vd = *(const float4*)&ytile[(lrA + 2) * PITCH + lb + 4 * w];
        yA[4*w+0]=va.x; yA[4*w+1]=va.y; yA[4*w+2]=va.z; yA[4*w+3]=va.w;
        yB[4*w+0]=vb.x; yB[4*w+1]=vb.y; yB[4*w+2]=vb.z; yB[4*w+3]=vb.w;
        yC[4*w+0]=vc.x; yC[4*w+1]=vc.y; yC[4*w+2]=vc.z; yC[4*w+3]=vc.w;
        yD[4*w+0]=vd.x; yD[4*w+1]=vd.y; yD[4*w+2]=vd.z; yD[4*w+3]=vd.w;
    }

    // ---------------- pooled 8x8 ratio term (from staged tile) -----------
    // This thread's 2x8 central patch lies entirely in pooled block
    // (blockIdx.y*2 + ty/4, blockIdx.x*32 + tx).
    psum[tid] = ((yB[4] + yB[5]) + (yB[6] + yB[7]))
              + ((yB[8] + yB[9]) + (yB[10] + yB[11]))
              + ((yC[4] + yC[5]) + (yC[6] + yC[7]))
              + ((yC[8] + yC[9]) + (yC[10] + yC[11]));
    __syncthreads();
    if (tid < 64) {
        int g  = tid >> 5;                    // pooled row group in tile
        int cx = tid & 31;                    // pooled col in tile
        float s = 0.0f;
        #pragma unroll
        for (int q = 0; q < 4; ++q) s += psum[(g * 4 + q) * 32 + cx];
        float mean = s * (1.0f / 64.0f);
        int pr = blockIdx.y * 2 + g;
        int pc = blockIdx.x * 32 + cx;
        size_t sidx = ((size_t)bb * PS + pr) * PS + pc;
        float ratio = (mask_inv[sidx] != 0) ? 1.0f
                                            : source[sidx] / (mean + EPSV);
        float tval = tptr[0];
        float wgt  = 1.0f / (1.0f + expf(-10.0f * (1.0f - tval)));
        rm1w[tid]  = wgt * (ratio - 1.0f);
    }
    __syncthreads();

    // ---------------- vertical flux: cv_r rows iA-1, iA, iA+1 -------------
    // cv_r = resize(cv,(2047,2048)): rows x=(r+0.5)*1024/2047-0.5,
    // cols exact x2 (weights 0.75/0.25). Clamped == jax edge handling.
    // tv(iA,·) is shared: outgoing flux of row iA, incoming flux of row iB.
    float xA = ((float)(iA - 1) + 0.5f) * SV - 0.5f;
    float xB = ((float)iA       + 0.5f) * SV - 0.5f;
    float xC = ((float)iB       + 0.5f) * SV - 0.5f;
    int   rA0 = (int)floorf(xA); float wA = xA - (float)rA0;
    int   rB0 = (int)floorf(xB); float wB = xB - (float)rB0;
    int   rC0 = (int)floorf(xC); float wC = xC - (float)rC0;
    int rA0c = min(max(rA0, 0), CVN - 1), rA1c = min(max(rA0 + 1, 0), CVN - 1);
    int rB0c = min(max(rB0, 0), CVN - 1), rB1c = min(max(rB0 + 1, 0), CVN - 1);
    int rC0c = min(max(rC0, 0), CVN - 1), rC1c = min(max(rC0 + 1, 0), CVN - 1);
    float gA = (iA >= 1)      ? LCONST : 0.0f;  // rv=iA-1 valid iff >= 0
    const float gB = LCONST;                    // rv=iA <= 2046 always
    float gC = (iB <= HH - 2) ? LCONST : 0.0f;  // rv=iB valid iff <= H-2

    // Taps cv[m-1 .. m+4]; inner taps m..m+3 always in [0,1023] (m in
    // [0,1020]); only first/last tap can clamp.
    const int m  = j0 >> 1;
    const int d0 = (m == 0)    ? 0 : -1;            // first tap offset
    const int d5 = (m >= 1020) ? (1023 - m) : 4;    // last tap offset
    const float* pA0 = cvb + (size_t)rA0c * CVN + m;
    const float* pA1 = cvb + (size_t)rA1c * CVN + m;
    const float* pB0 = cvb + (size_t)rB0c * CVN + m;
    const float* pB1 = cvb + (size_t)rB1c * CVN + m;
    const float* pC0 = cvb + (size_t)rC0c * CVN + m;
    const float* pC1 = cvb + (size_t)rC1c * CVN + m;

    float vA[6], vB[6], vC[6];  // vertical-lerped taps, prescaled by g (L or 0)
    #pragma unroll
    for (int kk = 0; kk < 6; ++kk) {
        int off = (kk == 0) ? d0 : (kk == 5) ? d5 : (kk - 1);
        float a0v = pA0[off], a1v = pA1[off];
        float b0v = pB0[off], b1v = pB1[off];
        float c0v = pC0[off], c1v = pC1[off];
        vA[kk] = gA * (a0v + wA * (a1v - a0v));
        vB[kk] = gB * (b0v + wB * (b1v - b0v));
        vC[kk] = gC * (c0v + wC * (c1v - c0v));
    }

    // y1 at cols j0-1 .. j0+8 for both rows (c = 2m-1+cc)
    float y1A[10], y1B[10];
    #pragma unroll
    for (int cc = 0; cc < 10; ++cc) {
        int   kk  = cc >> 1;
        float wlo = (cc & 1) ? 0.25f : 0.75f;
        float whi = 1.0f - wlo;
        float cAv = wlo * vA[kk] + whi * vA[kk + 1];
        float cBv = wlo * vB[kk] + whi * vB[kk + 1];
        float cCv = wlo * vC[kk] + whi * vC[kk + 1];
        float tvA = cAv * (yB[cc + 3] - yA[cc + 3]);
        float tvB = cBv * (yC[cc + 3] - yB[cc + 3]);   // shared flux
        float tvC = cCv * (yD[cc + 3] - yC[cc + 3]);
        y1A[cc] = yB[cc + 3] + tvB - tvA;
        y1B[cc] = yC[cc + 3] + tvC - tvB;
    }

    // ---------------- horizontal flux: th at cols j0-1 .. j0+7 ------------
    // Column coordinates shared by both rows. id needs no clamp:
    // monotonicity of u gives id >= 0; floor-difference bound gives
    // id <= 129, so id+1 <= 130 < NSEG.
    const float* csegA = &chseg[(2 * ty)     * NSEG];
    const float* csegB = &chseg[(2 * ty + 1) * NSEG];
    float thA[9], thB[9];
    #pragma unroll
    for (int cc = 0; cc < 9; ++cc) {
        float cf = (float)(j0 - 1 + cc);
        float u  = (cf + 0.5f) * SV - 0.5f;
        float uf = floorf(u);
        int  id  = (int)uf - kLo;
        float wc = u - uf;
        float a0 = csegA[id], a1 = csegA[id + 1];
        float b0 = csegB[id], b1 = csegB[id + 1];
        float chA = a0 + wc * (a1 - a0);               // L already folded in
        float chB = b0 + wc * (b1 - b0);
        thA[cc] = chA * (y1A[cc + 1] - y1A[cc]);
        thB[cc] = chB * (y1B[cc + 1] - y1B[cc]);
    }
    // Only possible out-of-range flux columns: c = -1 (j0==0, cc==0) and
    // c = W-1 (j0==W-8, cc==8).
    if (j0 == 0)      { thA[0] = 0.0f; thB[0] = 0.0f; }
    if (j0 == WW - 8) { thA[8] = 0.0f; thB[8] = 0.0f; }

    // ---------------- combine: dy + weight * y * (ratio - 1) --------------
    const float rw = rm1w[(ty >> 2) * 32 + tx];
    float oA[8], oB[8];
    #pragma unroll
    for (int c2 = 0; c2 < 8; ++c2) {
        float ycA = yB[c2 + 4];
        float ycB = yC[c2 + 4];
        oA[c2] = (y1A[c2 + 1] - ycA) + thA[c2 + 1] - thA[c2] + ycA * rw;
        oB[c2] = (y1B[c2 + 1] - ycB) + thB[c2 + 1] - thB[c2] + ycB * rw;
    }
    float* opA = out + ((size_t)bb * HH + iA) * WW + j0;
    float* opB = out + ((size_t)bb * HH + iB) * WW + j0;
    *(float4*)(opA)     = make_float4(oA[0], oA[1], oA[2], oA[3]);
    *(float4*)(opA + 4) = make_float4(oA[4], oA[5], oA[6], oA[7]);
    *(float4*)(opB)     = make_float4(oB[0], oB[1], oB[2], oB[3]);
    *(float4*)(opB + 4) = make_float4(oB[4], oB[5], oB[6], oB[7]);
}

extern "C" void kernel_launch(void* const* d_in, const int* in_sizes, int n_in,
                              void* d_out, int out_size, void* d_ws, size_t ws_size,
                              hipStream_t stream)
{
    // setup_inputs() order: t, y, cv, ch, source, mask_inv
    const float* t    = (const float*)d_in[0];
    const float* yin  = (const float*)d_in[1];
    const float* cv   = (const float*)d_in[2];
    const float* ch   = (const float*)d_in[3];
    const float* src  = (const float*)d_in[4];
    const int*   mask = (const int*)d_in[5];
    float* out = (float*)d_out;

    const int B = in_sizes[1] / (HH * WW);        // 4
    dim3 grid(WW / COLSZ, HH / TROWS, B);         // 8 x 128 x 4
    dim3 block(NTHR);
    diffusion_fused_kernel<<<grid, block, 0, stream>>>(t, yin, cv, ch, src, mask, out);

    (void)n_in; (void)out_size; (void)d_ws; (void)ws_size;
}